// WeightsLib2DLite_22222160789752
// MI455X (gfx1250) — compile-verified
//
#include <hip/hip_runtime.h>

// B=8, C=16, R=16, M=N=512.
// out = real part of  sum_c rot( Wi^T (wb*(1+norm(mi^T@mj))) Wj + td_c , rd_c )
// which collapses to: out = Ci*s1 - Cr*s0 + off  with 3 precomputed scalars.
// Complex GEMMs use Gauss 3-mult: P1=ArBr, P2=AiBi, P3=(Ar+Ai)(Br+Bi);
// Cr = P1-P2, Ci = P3-P1-P2.

#define Bsz 8
#define Csz 16
#define Rsz 16
#define Msz 512
#define Nsz 512

typedef __attribute__((ext_vector_type(2))) float v2f;
typedef __attribute__((ext_vector_type(8))) float v8f;

// f32 WMMA: D(16x16,f32) = A(16x4,f32) * B(4x16,f32) + C
__device__ __forceinline__ v8f wmma4(v2f a, v2f b, v8f c) {
  return __builtin_amdgcn_wmma_f32_16x16x4_f32(false, a, false, b, (short)0, c, false, false);
}

// ---------- workspace layout (floats) ----------
#define T_OFF      0                         // t[b][k][c][comp] : 8*4*16*2 = 1024
#define ST_OFF     1024                      // stats: sum[8], sumsq[8]
#define SC_OFF     1040                      // s0, s1, off
#define MISUM_OFF  2048                      // [B][R][M][2] = 131072
#define MJSUM_OFF  (MISUM_OFF + Bsz*Rsz*Msz*2)
#define MOD_OFF    (MJSUM_OFF + Bsz*Rsz*Nsz*2)          // [B][M][N][2] (reused as W8)
#define T1_OFF     (MOD_OFF + (size_t)Bsz*Msz*Nsz*2)    // [B][N][M][2]  (stored TRANSPOSED)

// ---------------- K1: tiny prep -----------------
__global__ __launch_bounds__(256)
void wl2d_prep_kernel(const float* __restrict__ x, const float* __restrict__ mt,
                      const float* __restrict__ td, const float* __restrict__ rd,
                      float* __restrict__ ws) {
  float* t = ws + T_OFF;
  int tid = threadIdx.x;
  for (int idx = tid; idx < Bsz * 4 * Csz * 2; idx += 256) {
    int comp = idx & 1;
    int c = (idx >> 1) & 15;
    int k = (idx >> 5) & 3;
    int b = idx >> 7;
    int l = 2 * c + comp;
    float s = 0.f;
#pragma unroll
    for (int cc = 0; cc < Csz; ++cc)
      s += x[b * Csz + cc] * mt[(k * Csz + cc) * (2 * Csz) + l];
    t[idx] = s;  // layout b*128 + k*32 + c*2 + comp
  }
  if (tid < 16) ws[ST_OFF + tid] = 0.f;  // zero batch stats each launch
  if (tid == 0) {
    float s0 = 0.f, s1 = 0.f, ofv = 0.f;
#pragma unroll
    for (int c = 0; c < Csz; ++c) {
      float r0 = rd[c * 2], r1 = rd[c * 2 + 1];
      float t0 = td[c * 2], t1 = td[c * 2 + 1];
      s0 += r0; s1 += r1;
      ofv += t1 * r1 - t0 * r0;
    }
    ws[SC_OFF + 0] = s0; ws[SC_OFF + 1] = s1; ws[SC_OFF + 2] = ofv;
  }
}

// ------- K2: rotated low-rank factors, reduced over C -------
__global__ __launch_bounds__(256)
void wl2d_modfactors_kernel(const float* __restrict__ mod_i, const float* __restrict__ mod_j,
                            const float* __restrict__ inv, float* __restrict__ ws) {
  const int total = Bsz * Rsz * Msz;  // 65536
  int g = blockIdx.x * blockDim.x + threadIdx.x;
  if (g >= 2 * total) return;
  bool isJ = g >= total;
  int p = isJ ? g - total : g;
  int m = p & (Msz - 1);
  int r = (p >> 9) & (Rsz - 1);
  int b = p >> 13;
  const float* src = isJ ? mod_j : mod_i;
  const float* tt = ws + T_OFF + b * 128 + (isJ ? 2 : 0) * 32;  // translate row k, rotate row k+1
  float acc0 = 0.f, acc1 = 0.f;
#pragma unroll
  for (int c = 0; c < Csz; ++c) {
    int si = ((c * Rsz + r) * Msz + m) * 2;
    float a0 = src[si + 0] + tt[c * 2 + 0];
    float a1 = src[si + 1] + tt[c * 2 + 1];
    float z0 = tt[32 + c * 2 + 0];
    float z1 = tt[32 + c * 2 + 1];
    float u0 = a1 * z1 - a0 * z0;   // _rot
    float u1 = a0 * z1 + a1 * z0;
    float i0 = inv[c * 2], i1 = inv[c * 2 + 1];
    if (!isJ) {
      acc0 += u1 * i1 - u0 * i0;    // _rot by inversions
      acc1 += u0 * i1 + u1 * i0;
    } else {
      float d = u0 * u0 + u1 * u1;  // division-like inversion
      acc0 += (i0 * u0 + i1 * u1) / d;
      acc1 += (i0 * u1 - i1 * u0) / d;
    }
  }
  float* dst = ws + (isJ ? MJSUM_OFF : MISUM_OFF);
  dst[p * 2 + 0] = acc0;
  dst[p * 2 + 1] = acc1;
}

// ------- K3: mod = mi^T @ mj  (complex, K=16) via f32 WMMA (Gauss) + stats -------
__global__ __launch_bounds__(32)
void wl2d_modgemm_kernel(float* __restrict__ ws) {
  const float* A = ws + MISUM_OFF + (size_t)blockIdx.z * Rsz * Msz * 2;   // [r][m][2]
  const float* Bm = ws + MJSUM_OFF + (size_t)blockIdx.z * Rsz * Nsz * 2;  // [r][n][2]
  float* modb = ws + MOD_OFF + (size_t)blockIdx.z * Msz * Nsz * 2;
  int lane = threadIdx.x, half = lane >> 4, lm = lane & 15;
  int m = blockIdx.x * 16 + lm;
  int n = blockIdx.y * 16 + lm;
  int klo = half * 2;
  v8f p1 = {}, p2 = {}, p3 = {};
#pragma unroll
  for (int k = 0; k < Rsz; k += 4) {
    v2f e0 = *(const v2f*)(A + ((k + klo) * Msz + m) * 2);       // mi^T: A[m,r]=misum[r][m]
    v2f e1 = *(const v2f*)(A + ((k + klo + 1) * Msz + m) * 2);
    v2f f0 = *(const v2f*)(Bm + ((k + klo) * Nsz + n) * 2);
    v2f f1 = *(const v2f*)(Bm + ((k + klo + 1) * Nsz + n) * 2);
    v2f ar = {e0.x, e1.x}, ai = {e0.y, e1.y}, as = ar + ai;
    v2f br = {f0.x, f1.x}, bi = {f0.y, f1.y}, bs = br + bi;
    p1 = wmma4(ar, br, p1);
    p2 = wmma4(ai, bi, p2);
    p3 = wmma4(as, bs, p3);
  }
  float lsum = 0.f, lsq = 0.f;
#pragma unroll
  for (int v = 0; v < 8; ++v) {
    int row = blockIdx.x * 16 + v + 8 * half;
    float cr = p1[v] - p2[v];
    float ci = p3[v] - p1[v] - p2[v];
    v2f o = {cr, ci};
    *(v2f*)(modb + ((size_t)row * Nsz + n) * 2) = o;
    lsum += cr; lsq += cr * cr;
  }
  atomicAdd(ws + ST_OFF + blockIdx.z, lsum);
  atomicAdd(ws + ST_OFF + 8 + blockIdx.z, lsq);
}

// ------- K4: normalize real part (ddof=1) and form W8 = wb*(1+mod), in place -------
__global__ __launch_bounds__(256)
void wl2d_w8_kernel(const float* __restrict__ wb, float* __restrict__ ws) {
  size_t g = (size_t)blockIdx.x * blockDim.x + threadIdx.x;  // B*M*N
  int b = (int)(g >> 18);
  size_t pos = g & (size_t)(Msz * Nsz - 1);
  const float cnt = (float)(Msz * Nsz);
  float mean = ws[ST_OFF + b] / cnt;
  float var = (ws[ST_OFF + 8 + b] - cnt * mean * mean) / (cnt - 1.f);
  float istd = rsqrtf(var);
  float* modp = ws + MOD_OFF + g * 2;
  float mrn = (modp[0] - mean) * istd;
  float mim = modp[1];
  modp[0] = wb[pos * 2 + 0] * (1.f + mrn);
  modp[1] = wb[pos * 2 + 1] * (1.f + mim);
}

// ------- K5: T1 = Wi^T @ W8  (complex 512^3, Gauss, 16x32 tile/wave) -------
// Output stored TRANSPOSED: T1t[n][m], so per-lane rows are contiguous and
// GEMM2's A-loads become coalesced [k][m] rows.
__global__ __launch_bounds__(32)
void wl2d_gemm1_kernel(const float* __restrict__ Wi, float* __restrict__ ws) {
  const float* W8b = ws + MOD_OFF + (size_t)blockIdx.z * Msz * Nsz * 2;
  float* T1b = ws + T1_OFF + (size_t)blockIdx.z * Msz * Nsz * 2;
  int lane = threadIdx.x, half = lane >> 4, lm = lane & 15;
  int m  = blockIdx.x * 16 + lm;
  int n0 = blockIdx.y * 32 + lm;
  int n1 = n0 + 16;
  int klo = half * 2;
  v8f p1a = {}, p2a = {}, p3a = {}, p1b = {}, p2b = {}, p3b = {};
  for (int k = 0; k < Msz; k += 4) {
    v2f e0 = *(const v2f*)(Wi + ((size_t)(k + klo) * Msz + m) * 2);      // A[m,k]=Wi[k,m]
    v2f e1 = *(const v2f*)(Wi + ((size_t)(k + klo + 1) * Msz + m) * 2);
    v2f ar = {e0.x, e1.x}, ai = {e0.y, e1.y}, as = ar + ai;
    v2f f0 = *(const v2f*)(W8b + ((size_t)(k + klo) * Nsz + n0) * 2);
    v2f f1 = *(const v2f*)(W8b + ((size_t)(k + klo + 1) * Nsz + n0) * 2);
    v2f g0 = *(const v2f*)(W8b + ((size_t)(k + klo) * Nsz + n1) * 2);
    v2f g1 = *(const v2f*)(W8b + ((size_t)(k + klo + 1) * Nsz + n1) * 2);
    v2f br0 = {f0.x, f1.x}, bi0 = {f0.y, f1.y}, bs0 = br0 + bi0;
    v2f br1 = {g0.x, g1.x}, bi1 = {g0.y, g1.y}, bs1 = br1 + bi1;
    p1a = wmma4(ar, br0, p1a);
    p2a = wmma4(ai, bi0, p2a);
    p3a = wmma4(as, bs0, p3a);
    p1b = wmma4(ar, br1, p1b);
    p2b = wmma4(ai, bi1, p2b);
    p3b = wmma4(as, bs1, p3b);
  }
  int rowbase = blockIdx.x * 16 + 8 * half;
  float* d0 = T1b + ((size_t)n0 * Msz + rowbase) * 2;  // 16 contiguous floats per lane
  float* d1 = T1b + ((size_t)n1 * Msz + rowbase) * 2;
#pragma unroll
  for (int v = 0; v < 8; ++v) {
    v2f oa = {p1a[v] - p2a[v], p3a[v] - p1a[v] - p2a[v]};
    *(v2f*)(d0 + v * 2) = oa;
    v2f ob = {p1b[v] - p2b[v], p3b[v] - p1b[v] - p2b[v]};
    *(v2f*)(d1 + v * 2) = ob;
  }
}

// ------- K6: W10 = T1 @ Wj (A read from transposed T1t), fused epilogue -------
__global__ __launch_bounds__(32)
void wl2d_gemm2_kernel(const float* __restrict__ Wj, const float* __restrict__ ws,
                       float* __restrict__ out) {
  const float* T1b = ws + T1_OFF + (size_t)blockIdx.z * Msz * Nsz * 2;  // T1t[n][m]
  float* outb = out + (size_t)blockIdx.z * Msz * Nsz;
  int lane = threadIdx.x, half = lane >> 4, lm = lane & 15;
  int m  = blockIdx.x * 16 + lm;
  int n0 = blockIdx.y * 32 + lm;
  int n1 = n0 + 16;
  int klo = half * 2;
  v8f p1a = {}, p2a = {}, p3a = {}, p1b = {}, p2b = {}, p3b = {};
  for (int k = 0; k < Nsz; k += 4) {
    v2f e0 = *(const v2f*)(T1b + ((size_t)(k + klo) * Msz + m) * 2);     // A[m,k]=T1t[k][m]
    v2f e1 = *(const v2f*)(T1b + ((size_t)(k + klo + 1) * Msz + m) * 2);
    v2f ar = {e0.x, e1.x}, ai = {e0.y, e1.y}, as = ar + ai;
    v2f f0 = *(const v2f*)(Wj + ((size_t)(k + klo) * Nsz + n0) * 2);
    v2f f1 = *(const v2f*)(Wj + ((size_t)(k + klo + 1) * Nsz + n0) * 2);
    v2f g0 = *(const v2f*)(Wj + ((size_t)(k + klo) * Nsz + n1) * 2);
    v2f g1 = *(const v2f*)(Wj + ((size_t)(k + klo + 1) * Nsz + n1) * 2);
    v2f br0 = {f0.x, f1.x}, bi0 = {f0.y, f1.y}, bs0 = br0 + bi0;
    v2f br1 = {g0.x, g1.x}, bi1 = {g0.y, g1.y}, bs1 = br1 + bi1;
    p1a = wmma4(ar, br0, p1a);
    p2a = wmma4(ai, bi0, p2a);
    p3a = wmma4(as, bs0, p3a);
    p1b = wmma4(ar, br1, p1b);
    p2b = wmma4(ai, bi1, p2b);
    p3b = wmma4(as, bs1, p3b);
  }
  float s0 = ws[SC_OFF + 0], s1 = ws[SC_OFF + 1], ofv = ws[SC_OFF + 2];
#pragma unroll
  for (int v = 0; v < 8; ++v) {
    int row = blockIdx.x * 16 + v + 8 * half;
    float cr = p1a[v] - p2a[v];
    float ci = p3a[v] - p1a[v] - p2a[v];
    outb[(size_t)row * Nsz + n0] = ci * s1 - cr * s0 + ofv;
    cr = p1b[v] - p2b[v];
    ci = p3b[v] - p1b[v] - p2b[v];
    outb[(size_t)row * Nsz + n1] = ci * s1 - cr * s0 + ofv;
  }
}

extern "C" void kernel_launch(void* const* d_in, const int* in_sizes, int n_in,
                              void* d_out, int out_size, void* d_ws, size_t ws_size,
                              hipStream_t stream) {
  const float* x   = (const float*)d_in[0];
  const float* Wi  = (const float*)d_in[1];
  const float* Wj  = (const float*)d_in[2];
  const float* wb  = (const float*)d_in[3];
  // d_in[4] translate_base, d_in[5] rotate_base: dead code in reference
  const float* td  = (const float*)d_in[6];
  const float* rd  = (const float*)d_in[7];
  const float* mi  = (const float*)d_in[8];
  const float* mj  = (const float*)d_in[9];
  const float* inv = (const float*)d_in[10];
  const float* mt  = (const float*)d_in[11];
  float* ws = (float*)d_ws;
  float* out = (float*)d_out;

  wl2d_prep_kernel<<<1, 256, 0, stream>>>(x, mt, td, rd, ws);

  int tot2 = 2 * Bsz * Rsz * Msz;  // 131072
  wl2d_modfactors_kernel<<<(tot2 + 255) / 256, 256, 0, stream>>>(mi, mj, inv, ws);

  dim3 gmod(Msz / 16, Nsz / 16, Bsz);  // one wave per 16x16 tile, K=16
  wl2d_modgemm_kernel<<<gmod, 32, 0, stream>>>(ws);

  int totE = Bsz * Msz * Nsz;  // 2097152
  wl2d_w8_kernel<<<totE / 256, 256, 0, stream>>>(wb, ws);

  dim3 ggemm(Msz / 16, Nsz / 32, Bsz);  // one wave per 16x32 complex tile
  wl2d_gemm1_kernel<<<ggemm, 32, 0, stream>>>(Wi, ws);
  wl2d_gemm2_kernel<<<ggemm, 32, 0, stream>>>(Wj, ws, out);
}